// Attention_13314398617962
// MI455X (gfx1250) — compile-verified
//
#include <hip/hip_runtime.h>
#include <math.h>

typedef float v2f __attribute__((ext_vector_type(2)));
typedef float v8f __attribute__((ext_vector_type(8)));

#define BN_EPS 1e-5f

// dims
#define T 512
#define B 64
#define HU 1024
#define NROWS (T * B)          // 32768, flat row = t*64 + b  (== h's leading layout)
#define TCHUNKS 4              // t-split for the weighted-sum pass

// d_ws float layout:
//   [0, 16384)        W0pair : W0[1024+k][n] padded to n=0..15, k-paired layout
//                              Wp[((k>>1)*16 + n)*2 + (k&1)]
//   [16384, 17408)    sp     : per-b si-projection + b0, 64 x 16
//   [17408, 50176)    scores : 32768 flat scores
//   [50176, 50688)    wts    : 512 softmax weights
//   [50688, 312832)   part   : TCHUNKS x (64*1024) partial weighted sums
#define WS_W0PAIR 0
#define WS_SP     16384
#define WS_SCORE  17408
#define WS_WTS    50176
#define WS_PART   50688

// ---------------- kernel 1: pad + pair W0's h-half ----------------------------
// Paired layout puts the (k, k+1) values a lane needs adjacent -> one b64 load.
__global__ void prep_w0pair_kernel(const float* __restrict__ W0,
                                   float* __restrict__ W0pair) {
    int idx = blockIdx.x * blockDim.x + threadIdx.x;   // 0..16383
    int k = idx >> 4, n = idx & 15;
    float v = (n < 10) ? W0[(HU + k) * 10 + n] : 0.0f;
    W0pair[(((k >> 1) * 16 + n) << 1) + (k & 1)] = v;
}

// ---------------- kernel 2: sp[b][n] = b0[n] + si[b,:] @ W0[:1024, n] ----------
__global__ void si_proj_kernel(const float* __restrict__ si,
                               const float* __restrict__ W0,
                               const float* __restrict__ b0,
                               float* __restrict__ sp) {
    __shared__ float red[256 * 10];
    int bi = blockIdx.x, tid = threadIdx.x;
    float p[10];
#pragma unroll
    for (int n = 0; n < 10; ++n) p[n] = 0.0f;
    for (int k = tid; k < HU; k += 256) {
        float sv = si[bi * HU + k];
        const float* w = W0 + k * 10;
#pragma unroll
        for (int n = 0; n < 10; ++n) p[n] += sv * w[n];
    }
#pragma unroll
    for (int n = 0; n < 10; ++n) red[tid * 10 + n] = p[n];
    __syncthreads();
    for (int s = 128; s > 0; s >>= 1) {
        if (tid < s) {
#pragma unroll
            for (int n = 0; n < 10; ++n) red[tid * 10 + n] += red[(tid + s) * 10 + n];
        }
        __syncthreads();
    }
    if (tid < 10) sp[bi * 16 + tid] = red[tid] + b0[tid];
}

// ---------------- kernel 3: WMMA projection + MLP tail -> scores ---------------
// block = 256 threads = 8 waves; each wave owns a 16-row tile (rows = flat t*64+b)
__global__ void __launch_bounds__(256)
scores_kernel(const float* __restrict__ h,
              const float* __restrict__ W0pair,
              const float* __restrict__ sp,
              const float* __restrict__ g0, const float* __restrict__ be0,
              const float* __restrict__ m0, const float* __restrict__ v0,
              const float* __restrict__ W1, const float* __restrict__ b1,
              const float* __restrict__ g1, const float* __restrict__ be1,
              const float* __restrict__ m1, const float* __restrict__ v1,
              const float* __restrict__ W2, const float* __restrict__ b2,
              float* __restrict__ scores) {
    __shared__ float xt[8 * 16 * 16];   // per-wave 16x16 activation tiles

    const int tid  = threadIdx.x;
    const int wave = tid >> 5;
    const int lane = tid & 31;
    const int wavebase = (blockIdx.x * 8 + wave) * 16;

    const int n    = lane & 15;         // A: row-in-tile; B/C: column
    const int half = lane >> 4;         // upper half-wave handles K+2 / M+8

    // A fragment: h row (wavebase+n), K pair {k0+2*half, k0+2*half+1} -> b64 load
    const float* aptr = h + (size_t)(wavebase + n) * HU + 2 * half;
    // B fragment: paired W0; element ((k0/2 + half)*16 + n)*2 {+0,+1} -> b64 load
    const float* bptr = W0pair + half * 32 + n * 2;

    v8f acc = {};
#pragma unroll 4
    for (int k0 = 0; k0 < HU; k0 += 4) {
        v2f a = *(const v2f*)(aptr + k0);        // 8B-aligned global_load_b64
        v2f b = *(const v2f*)(bptr + k0 * 16);   // 8B-aligned global_load_b64
        // D = A(16x4) * B(4x16) + C  — f32 WMMA
        acc = __builtin_amdgcn_wmma_f32_16x16x4_f32(
            false, a, false, b, (short)0, acc, false, false);
    }

    // Epilogue part 1: add si-projection, fused BN0 (+ReLU), write 16x16 tile.
    // Lane owns column n, rows j + 8*half (C/D layout).
    if (n < 10) {
        float mg  = g0[n] * rsqrtf(v0[n] + BN_EPS);
        float off = be0[n] - m0[n] * mg;
#pragma unroll
        for (int j = 0; j < 8; ++j) {
            int m  = j + 8 * half;
            int bi = (wavebase + m) & 63;
            float z = acc[j] + sp[bi * 16 + n];
            z = fmaxf(z * mg + off, 0.0f);
            xt[(wave * 16 + m) * 16 + n] = z;
        }
    }
    __syncthreads();

    // Epilogue part 2: 16 lanes, one row each: 10 -> 5 (BN1+ReLU) -> 1
    if (lane < 16) {
        const int m   = lane;
        const int row = wavebase + m;
        const float* x = &xt[(wave * 16 + m) * 16];
        float sc = b2[0];
#pragma unroll
        for (int c = 0; c < 5; ++c) {
            float y = b1[c];
#pragma unroll
            for (int q = 0; q < 10; ++q) y += x[q] * W1[q * 5 + c];
            float mg1 = g1[c] * rsqrtf(v1[c] + BN_EPS);
            y = fmaxf(y * mg1 + (be1[c] - m1[c] * mg1), 0.0f);
            sc += y * W2[c];
        }
        scores[row] = sc;
    }
}

// ---------------- kernel 4: global softmax over 32768, keep first 512 ----------
__global__ void softmax_kernel(const float* __restrict__ scores,
                               float* __restrict__ wts) {
    __shared__ float red[1024];
    int tid = threadIdx.x;
    float mx = -INFINITY;
    for (int i = tid; i < NROWS; i += 1024) mx = fmaxf(mx, scores[i]);
    red[tid] = mx;
    __syncthreads();
    for (int s = 512; s > 0; s >>= 1) {
        if (tid < s) red[tid] = fmaxf(red[tid], red[tid + s]);
        __syncthreads();
    }
    float gmax = red[0];
    __syncthreads();
    float sum = 0.0f;
    for (int i = tid; i < NROWS; i += 1024) sum += __expf(scores[i] - gmax);
    red[tid] = sum;
    __syncthreads();
    for (int s = 512; s > 0; s >>= 1) {
        if (tid < s) red[tid] += red[tid + s];
        __syncthreads();
    }
    float gsum = red[0];
    if (tid < T) wts[tid] = __expf(scores[tid] - gmax) / gsum;
}

// ---------------- kernel 5a: partial ci over a t-chunk -------------------------
// grid (2, 64, TCHUNKS): x = k-half, y = b, z = t-chunk; 128 thr x float4 = 512 k
__global__ void ci_partial_kernel(const float* __restrict__ h,
                                  const float* __restrict__ wts,
                                  float* __restrict__ part) {
    int bi = blockIdx.y;
    int tc = blockIdx.z;
    int k0 = blockIdx.x * 512 + threadIdx.x * 4;
    const int tlen = T / TCHUNKS;               // 128
    const float* hp = h + (size_t)(tc * tlen) * (B * HU) + (size_t)bi * HU + k0;
    const float* wp = wts + tc * tlen;
    float4 acc = make_float4(0.f, 0.f, 0.f, 0.f);
#pragma unroll 8
    for (int t = 0; t < tlen; ++t) {
        float w = wp[t];
        float4 hv = *(const float4*)(hp + (size_t)t * (B * HU));
        acc.x += w * hv.x;
        acc.y += w * hv.y;
        acc.z += w * hv.z;
        acc.w += w * hv.w;
    }
    *(float4*)(part + (size_t)tc * (B * HU) + bi * HU + k0) = acc;
}

// ---------------- kernel 5b: reduce TCHUNKS partials (deterministic order) -----
__global__ void ci_reduce_kernel(const float* __restrict__ part,
                                 float* __restrict__ out) {
    int idx = (blockIdx.x * blockDim.x + threadIdx.x) * 4;   // 0..65532
    float4 acc = *(const float4*)(part + idx);
#pragma unroll
    for (int c = 1; c < TCHUNKS; ++c) {
        float4 p = *(const float4*)(part + (size_t)c * (B * HU) + idx);
        acc.x += p.x; acc.y += p.y; acc.z += p.z; acc.w += p.w;
    }
    *(float4*)(out + idx) = acc;
}

extern "C" void kernel_launch(void* const* d_in, const int* in_sizes, int n_in,
                              void* d_out, int out_size, void* d_ws, size_t ws_size,
                              hipStream_t stream) {
    const float* si  = (const float*)d_in[0];
    const float* h   = (const float*)d_in[1];
    const float* W0  = (const float*)d_in[2];
    const float* b0  = (const float*)d_in[3];
    const float* g0  = (const float*)d_in[4];
    const float* be0 = (const float*)d_in[5];
    const float* m0  = (const float*)d_in[6];
    const float* v0  = (const float*)d_in[7];
    const float* W1  = (const float*)d_in[8];
    const float* b1  = (const float*)d_in[9];
    const float* g1  = (const float*)d_in[10];
    const float* be1 = (const float*)d_in[11];
    const float* m1  = (const float*)d_in[12];
    const float* v1  = (const float*)d_in[13];
    const float* W2  = (const float*)d_in[14];
    const float* b2  = (const float*)d_in[15];

    float* ws     = (float*)d_ws;
    float* W0pair = ws + WS_W0PAIR;
    float* sp     = ws + WS_SP;
    float* scores = ws + WS_SCORE;
    float* wts    = ws + WS_WTS;
    float* part   = ws + WS_PART;
    float* out    = (float*)d_out;

    prep_w0pair_kernel<<<64, 256, 0, stream>>>(W0, W0pair);
    si_proj_kernel<<<64, 256, 0, stream>>>(si, W0, b0, sp);
    scores_kernel<<<NROWS / 128, 256, 0, stream>>>(
        h, W0pair, sp, g0, be0, m0, v0, W1, b1, g1, be1, m1, v1, W2, b2, scores);
    softmax_kernel<<<1, 1024, 0, stream>>>(scores, wts);
    dim3 g5(2, 64, TCHUNKS);
    ci_partial_kernel<<<g5, 128, 0, stream>>>(h, wts, part);
    ci_reduce_kernel<<<64, 256, 0, stream>>>(part, out);
}